// Attention_17489106830121
// MI455X (gfx1250) — compile-verified
//
#include <hip/hip_runtime.h>
#include <math.h>
#include <stdint.h>

// ---------------------------------------------------------------------------
// PVT-style spatial-reduction attention, algebraically collapsed:
//   qs[b,h,n]   = x[b,n,:] . wq_sum[:,h]
//   conv rows   = Patch[2048,4096] @ W2[4096,256] + bsr   (the only big GEMM)
//   s_k/s_v     = LN-folded dot products of conv rows
//   w[b,h,n]    = sum_k softmax_k(qs*Lk[h]*SCALE * s_k[b,k]) * s_v[b,k]
//   out[b,n,:]  = w[b,:,n] @ M[8,256] + bproj
// Conv GEMM on V_WMMA_F32_16X16X4_F32; tile staging via async-to-LDS engine
// (global_load_async_to_lds_b128 + s_wait_asynccnt), double buffered.
// ---------------------------------------------------------------------------

typedef __attribute__((ext_vector_type(2))) float v2f;
typedef __attribute__((ext_vector_type(8))) float v8f;

#define BATCH   8
#define NTOK    4096
#define CCH     256
#define N2TOK   256
#define EPSV    1e-5f
#define SCALEV  0.17677669529663687f   // 32^-0.5

// workspace layout (float offsets)
#define WS_W2    0                      // [4096][256] repacked conv weight
#define WS_SK    1048576                // [B][256]
#define WS_SV    (WS_SK + 2048)         // [B][256]
#define WS_WQS   (WS_SV + 2048)         // [256][8]  wq_sum[i][h]
#define WS_GK    (WS_WQS + 2048)        // [256] gamma*wk_sum
#define WS_GV    (WS_GK + 256)          // [256] gamma*wv_sum
#define WS_SCAL  (WS_GV + 256)          // Gk,Gv,Bk,Bv
#define WS_LK    (WS_SCAL + 8)          // [8]
#define WS_M     (WS_LK + 8)            // [8][256]

// ---- CDNA5 async copy helpers (inline asm: portable across toolchains) ----
__device__ __forceinline__ void async_load_b128(uint32_t lds_off,
                                                const float* gptr) {
  // per-lane: copies 16 bytes from the global address gptr to the LDS byte offset lds_off, tracked by ASYNCcnt
  asm volatile("global_load_async_to_lds_b128 %0, %1, off"
               :: "v"(lds_off), "v"(gptr)
               : "memory");
}
__device__ __forceinline__ void wait_async0() {
  asm volatile("s_wait_asynccnt 0x0" ::: "memory");
}
__device__ __forceinline__ uint32_t lds_off32(const void* p) {
  // generic LDS pointer: high bits = shared aperture, low 32 = LDS byte offset
  return (uint32_t)(uintptr_t)p;
}

// ---------------------------------------------------------------------------
__global__ __launch_bounds__(256) void prep_kernel(
    const float* __restrict__ Wq, const float* __restrict__ Wkv,
    const float* __restrict__ gamma, const float* __restrict__ beta,
    const float* __restrict__ Wproj, const float* __restrict__ k_learn,
    const float* __restrict__ v_learn, float* __restrict__ ws) {
  __shared__ float rk[256], rv[256], rbk[256], rbv[256];
  int i = threadIdx.x;

  // wk_sum[i] = sum_j Wkv[i][2j]  (k cols), wv_sum = odd cols (v cols)
  float wk = 0.f, wv = 0.f;
  const float* row = Wkv + (size_t)i * 512;
  for (int j = 0; j < 256; ++j) { wk += row[2 * j]; wv += row[2 * j + 1]; }
  float gki = gamma[i] * wk, gvi = gamma[i] * wv;
  ws[WS_GK + i] = gki;
  ws[WS_GV + i] = gvi;
  rk[i] = gki; rv[i] = gvi; rbk[i] = beta[i] * wk; rbv[i] = beta[i] * wv;

  // wq_sum[i][h] = sum_ch Wq[i][h*32+ch]
  const float* qrow = Wq + (size_t)i * 256;
  for (int h = 0; h < 8; ++h) {
    float s = 0.f;
    for (int ch = 0; ch < 32; ++ch) s += qrow[h * 32 + ch];
    ws[WS_WQS + i * 8 + h] = s;
  }
  // M[d][o] = sum_ch lv[d][ch] * Wproj[ch*8+d][o]   (o = this thread)
  for (int d = 0; d < 8; ++d) {
    float s = 0.f;
    for (int ch = 0; ch < 32; ++ch)
      s += v_learn[d * 32 + ch] * Wproj[(size_t)(ch * 8 + d) * 256 + i];
    ws[WS_M + d * 256 + i] = s;
  }
  if (i < 8) {
    float s = 0.f;
    for (int q = 0; q < 32; ++q) s += k_learn[i * 32 + q];
    ws[WS_LK + i] = s;
  }
  __syncthreads();
  for (int st = 128; st > 0; st >>= 1) {
    if (i < st) { rk[i] += rk[i + st]; rv[i] += rv[i + st];
                  rbk[i] += rbk[i + st]; rbv[i] += rbv[i + st]; }
    __syncthreads();
  }
  if (i == 0) {
    ws[WS_SCAL + 0] = rk[0];  // Gk = sum gamma*wk_sum
    ws[WS_SCAL + 1] = rv[0];  // Gv
    ws[WS_SCAL + 2] = rbk[0]; // Bk = beta . wk_sum
    ws[WS_SCAL + 3] = rbv[0]; // Bv
  }
}

// ---------------------------------------------------------------------------
// Repack Wsr (OIHW [256,256,4,4]) -> W2[k][o], k = (kh*4+kw)*256 + i
__global__ __launch_bounds__(256) void wsr_transpose_kernel(
    const float* __restrict__ Wsr, float* __restrict__ ws) {
  int idx = blockIdx.x * 256 + threadIdx.x;   // 0 .. 1048575
  int o = idx & 255;
  int k = idx >> 8;
  int i = k & 255;
  int khkw = k >> 8;
  int kh = khkw >> 2, kw = khkw & 3;
  ws[WS_W2 + idx] = Wsr[(((size_t)o * 256 + i) * 4 + kh) * 4 + kw];
}

// ---------------------------------------------------------------------------
// Conv GEMM (WMMA fp32) + LayerNorm-folded row reductions -> s_k, s_v.
// One block = 16 conv-output rows; 8 waves each own 32 output columns.
// A tiles are fetched by the async engine, double-buffered across (kh,kw)
// segments so the copy of segment kb+1 overlaps the 64 WMMA steps of kb.
#define ALS 260   // A tile row stride: 1040B (16B-aligned rows, conflict-free)
#define XTS 264
__global__ __launch_bounds__(256) void conv_ln_kernel(
    const float* __restrict__ x, const float* __restrict__ bsr,
    float* __restrict__ ws) {
  __shared__ float A[2][16 * ALS];
  __shared__ float xt[16 * XTS];
  __shared__ float gkl[256], gvl[256], bsl[256];
  __shared__ float red[16 * 16 * 4];

  const float* W2 = ws + WS_W2;
  int t = threadIdx.x;
  int lane = t & 31, wv = t >> 5;
  int ln16 = lane & 15;
  int koff = (lane >> 4) << 1;      // 0 for lanes 0-15, 2 for 16-31 (A/B striping)
  int n0 = wv * 32;

  int R0 = blockIdx.x * 16;         // global conv row tile (B*N2 = 2048 rows)
  int b = R0 >> 8;
  int n2base = R0 & 255;

  gkl[t] = ws[WS_GK + t];
  gvl[t] = ws[WS_GV + t];
  bsl[t] = bsr[t];

  v8f acc0 = {}; v8f acc1 = {};

  // async-stage one (kh,kw) segment: 16 rows x 256 floats, b128 per lane
  auto stage = [&](int kb, int buf) {
    int kh = kb >> 2, kw = kb & 3;
    for (int q = 0; q < 4; ++q) {
      int c = q * 256 + t;            // 1024 chunks of 16B
      int m = c >> 6;                 // row 0..15
      int col = (c & 63) << 2;        // col 0,4,...,252
      int n2 = n2base + m;
      int h2 = n2 >> 4, w2 = n2 & 15;
      int pix = (h2 * 4 + kh) * 64 + (w2 * 4 + kw);
      const float* g = x + ((size_t)b * NTOK + pix) * 256 + col;
      async_load_b128(lds_off32(&A[buf][m * ALS + col]), g);
    }
  };

  stage(0, 0);
  for (int kb = 0; kb < 16; ++kb) {           // (kh,kw) segments of the patch
    const float* Ab = A[kb & 1];
    wait_async0();                  // this wave's staged chunks are in LDS
    __syncthreads();                // everyone's chunks visible; old buf dead
    if (kb < 15) stage(kb + 1, (kb + 1) & 1);
    const float* Wb = W2 + (size_t)kb * 256 * 256;
    // pull next W2 segment toward the caches while we compute on this one
    __builtin_prefetch(W2 + (size_t)((kb + 1) & 15) * 256 * 256 + t * 64, 0, 0);
    for (int ks = 0; ks < 64; ++ks) {
      int k0 = ks * 4;
      // A frag 16x4: lane l<16 -> M=l,K={k0,k0+1}; lane>=16 -> M=l-16,K={k0+2,k0+3}
      v2f a;
      a.x = Ab[ln16 * ALS + k0 + koff];
      a.y = Ab[ln16 * ALS + k0 + koff + 1];
      // B frag 4x16: VGPR0 = K {0|2}, VGPR1 = K {1|3}; N = n-tile col = lane%16
      const float* Wp = Wb + (size_t)(k0 + koff) * 256;
      v2f b0, b1;
      b0.x = Wp[n0 + ln16];        b0.y = Wp[256 + n0 + ln16];
      b1.x = Wp[n0 + 16 + ln16];   b1.y = Wp[256 + n0 + 16 + ln16];
      acc0 = __builtin_amdgcn_wmma_f32_16x16x4_f32(false, a, false, b0,
                                                   (short)0, acc0, false, false);
      acc1 = __builtin_amdgcn_wmma_f32_16x16x4_f32(false, a, false, b1,
                                                   (short)0, acc1, false, false);
    }
  }
  __syncthreads();
  // spill D tiles: VGPR r -> row r (lanes 0-15) / r+8 (lanes 16-31), col = lane%16
  int mb = (lane < 16) ? 0 : 8;
  for (int r = 0; r < 8; ++r) {
    xt[(mb + r) * XTS + n0 + ln16]      = acc0[r];
    xt[(mb + r) * XTS + n0 + 16 + ln16] = acc1[r];
  }
  __syncthreads();
  // per-row {sum, sumsq, dot gk, dot gv} with conv bias applied
  {
    int m = t >> 4, p = t & 15;
    float s1 = 0.f, s2 = 0.f, s3 = 0.f, s4 = 0.f;
    for (int c0 = 0; c0 < 16; ++c0) {
      int c = p * 16 + c0;
      float val = xt[m * XTS + c] + bsl[c];
      s1 += val; s2 += val * val; s3 += val * gkl[c]; s4 += val * gvl[c];
    }
    red[(m * 16 + p) * 4 + 0] = s1; red[(m * 16 + p) * 4 + 1] = s2;
    red[(m * 16 + p) * 4 + 2] = s3; red[(m * 16 + p) * 4 + 3] = s4;
  }
  __syncthreads();
  if (t < 16) {
    float s1 = 0.f, s2 = 0.f, s3 = 0.f, s4 = 0.f;
    for (int p = 0; p < 16; ++p) {
      s1 += red[(t * 16 + p) * 4 + 0]; s2 += red[(t * 16 + p) * 4 + 1];
      s3 += red[(t * 16 + p) * 4 + 2]; s4 += red[(t * 16 + p) * 4 + 3];
    }
    float Gk = ws[WS_SCAL + 0], Gv = ws[WS_SCAL + 1];
    float Bk = ws[WS_SCAL + 2], Bv = ws[WS_SCAL + 3];
    float mu  = s1 * (1.0f / 256.0f);
    float var = s2 * (1.0f / 256.0f) - mu * mu;
    float inv = rsqrtf(var + EPSV);
    int n2 = n2base + t;
    ws[WS_SK + b * 256 + n2] = inv * (s3 - mu * Gk) + Bk;
    ws[WS_SV + b * 256 + n2] = inv * (s4 - mu * Gv) + Bv;
  }
}

// ---------------------------------------------------------------------------
// Fused qs + stable softmax over 256 keys + rank-8 output projection.
// x rows staged via the async-to-LDS engine while scalars load normally.
#define XRS 260   // 1040B row stride: 16B aligned, 4 distinct banks per wave
__global__ __launch_bounds__(256) void attn_out_kernel(
    const float* __restrict__ x, const float* __restrict__ bproj,
    const float* __restrict__ ws, float* __restrict__ out) {
  __shared__ float xr[32 * XRS];
  __shared__ float wqs[2048];
  __shared__ float skl[256], svl[256];
  __shared__ float Ml[2048];
  __shared__ float bpl[256];
  __shared__ float warr[32 * 8];
  __shared__ float Lkl[8];
  __shared__ float mxmn[2];

  int t = threadIdx.x;
  int b = blockIdx.x >> 7;        // 128 chunks of 32 tokens per batch
  int chunk = blockIdx.x & 127;
  int nbase = chunk * 32;

  // async-stage 32 rows x 256 floats (2048 x 16B chunks, 8 per thread)
  for (int q = 0; q < 8; ++q) {
    int c = q * 256 + t;
    int m = c >> 6;                 // row 0..31
    int col = (c & 63) << 2;        // col 0,4,...,252
    const float* g = x + ((size_t)b * NTOK + nbase + m) * 256 + col;
    async_load_b128(lds_off32(&xr[m * XRS + col]), g);
  }
  for (int j = 0; j < 8; ++j) {
    wqs[j * 256 + t] = ws[WS_WQS + j * 256 + t];  // flat copy: index i*8+h
    Ml[j * 256 + t]  = ws[WS_M + j * 256 + t];    // flat copy: index d*256+o
  }
  skl[t] = ws[WS_SK + b * 256 + t];
  svl[t] = ws[WS_SV + b * 256 + t];
  bpl[t] = bproj[t];
  if (t < 8) Lkl[t] = ws[WS_LK + t];
  wait_async0();
  __syncthreads();
  if (t == 0) {
    float mx = skl[0], mn = skl[0];
    for (int k = 1; k < 256; ++k) { mx = fmaxf(mx, skl[k]); mn = fminf(mn, skl[k]); }
    mxmn[0] = mx; mxmn[1] = mn;
  }
  __syncthreads();
  {
    int nl = t >> 3, h = t & 7;
    float q = 0.f;
    for (int i = 0; i < 256; ++i) q += xr[nl * XRS + i] * wqs[i * 8 + h];
    float ts = q * Lkl[h] * SCALEV;
    float mref = (ts >= 0.f) ? ts * mxmn[0] : ts * mxmn[1];  // max_k(ts*s_k)
    float den = 0.f, num = 0.f;
    for (int k = 0; k < 256; ++k) {
      float e = __expf(ts * skl[k] - mref);
      den += e; num += e * svl[k];
    }
    warr[nl * 8 + h] = num / den;
  }
  __syncthreads();
  for (int nl = 0; nl < 32; ++nl) {
    float v = bpl[t];
    for (int d = 0; d < 8; ++d) v += warr[nl * 8 + d] * Ml[d * 256 + t];
    out[((size_t)b * NTOK + nbase + nl) * 256 + t] = v;
  }
}

// ---------------------------------------------------------------------------
extern "C" void kernel_launch(void* const* d_in, const int* in_sizes, int n_in,
                              void* d_out, int out_size, void* d_ws, size_t ws_size,
                              hipStream_t stream) {
  (void)in_sizes; (void)n_in; (void)out_size; (void)ws_size;
  const float* x       = (const float*)d_in[0];
  const float* Wq      = (const float*)d_in[1];
  const float* Wkv     = (const float*)d_in[2];
  const float* Wsr     = (const float*)d_in[3];
  const float* bsr     = (const float*)d_in[4];
  const float* gamma   = (const float*)d_in[5];
  const float* beta    = (const float*)d_in[6];
  const float* Wproj   = (const float*)d_in[7];
  const float* bproj   = (const float*)d_in[8];
  const float* k_learn = (const float*)d_in[9];
  const float* v_learn = (const float*)d_in[10];
  float* out = (float*)d_out;
  float* ws  = (float*)d_ws;

  hipLaunchKernelGGL(prep_kernel, dim3(1), dim3(256), 0, stream,
                     Wq, Wkv, gamma, beta, Wproj, k_learn, v_learn, ws);
  hipLaunchKernelGGL(wsr_transpose_kernel, dim3(4096), dim3(256), 0, stream, Wsr, ws);
  hipLaunchKernelGGL(conv_ln_kernel, dim3(128), dim3(256), 0, stream, x, bsr, ws);
  hipLaunchKernelGGL(attn_out_kernel, dim3(1024), dim3(256), 0, stream,
                     x, bproj, ws, out);
}